// SortClusterActQuant_68539088109686
// MI455X (gfx1250) — compile-verified
//
#include <hip/hip_runtime.h>
#include <stdint.h>

// C = 4096 channels, GROUP = 64, QMAX = 127
#define CHN   4096
#define CHN4  (CHN / 4)
#define QMAXF 127.0f

typedef float f4 __attribute__((ext_vector_type(4)));

// ---------------------------------------------------------------------------
// Kernel 1: fold the permutation into a per-channel scale LUT.
//   scale[c]     = group_scales[inv_perm[c] >> 6]
//   inv_scale[c] = 1 / scale[c]
// ---------------------------------------------------------------------------
__global__ void build_scale_lut_kernel(const float* __restrict__ group_scales,
                                       const int* __restrict__ inv_perm,
                                       float* __restrict__ scale_lut,
                                       float* __restrict__ inv_lut) {
    int c = blockIdx.x * blockDim.x + threadIdx.x;
    if (c < CHN) {
        float s = group_scales[inv_perm[c] >> 6];
        scale_lut[c] = s;
        inv_lut[c]   = 1.0f / s;
    }
}

// ---------------------------------------------------------------------------
// Kernel 2: streaming fake-quant.
//   out = clamp(rint(x * inv_scale[c]), -127, 127) * scale[c]
// Scale LUTs (2 x 16 KB) staged into LDS via CDNA5 async global->LDS copy.
// Bulk tensor streamed with 128-bit non-temporal loads/stores + prefetch.
// ---------------------------------------------------------------------------
__global__ void __launch_bounds__(256)
fakequant_kernel(const f4* __restrict__ x,
                 const float* __restrict__ scale_lut,
                 const float* __restrict__ inv_lut,
                 f4* __restrict__ out,
                 long long nvec) {
    __shared__ float s_scale[CHN];
    __shared__ float s_inv[CHN];

    // --- async copy both 16 KB LUTs into LDS (CDNA5 ASYNCcnt path) ---------
    // 256 threads x 4 iters x 16 B = 16 KB per table.
    #pragma unroll
    for (int i = 0; i < 4; ++i) {
        unsigned fidx  = (threadIdx.x + i * 256u) * 4u;   // float index
        unsigned boff  = fidx * 4u;                       // byte offset
        unsigned lds_s = (unsigned)(uintptr_t)&s_scale[fidx];
        unsigned lds_i = (unsigned)(uintptr_t)&s_inv[fidx];
        asm volatile("global_load_async_to_lds_b128 %0, %1, %2"
                     :: "v"(lds_s), "v"(boff), "s"(scale_lut) : "memory");
        asm volatile("global_load_async_to_lds_b128 %0, %1, %2"
                     :: "v"(lds_i), "v"(boff), "s"(inv_lut) : "memory");
    }
    asm volatile("s_wait_asynccnt 0" ::: "memory");
    __syncthreads();

    // --- streaming grid-stride loop over float4 elements -------------------
    long long v      = (long long)blockIdx.x * blockDim.x + threadIdx.x;
    long long stride = (long long)gridDim.x * blockDim.x;

    for (; v < nvec; v += stride) {
        // prefetch next stripe (gfx1250 global_prefetch_b8)
        if (v + stride < nvec)
            __builtin_prefetch(&x[v + stride], 0, 1);

        int cb = (int)(v & (CHN4 - 1)) << 2;              // channel base (0..4092)

        f4 xv = __builtin_nontemporal_load(&x[v]);        // streaming load (NT)
        f4 sv = *(const f4*)&s_scale[cb];                  // LDS ds_load_b128
        f4 iv = *(const f4*)&s_inv[cb];

        f4 y;
        #pragma unroll
        for (int k = 0; k < 4; ++k) {
            float q = __builtin_rintf(xv[k] * iv[k]);     // v_rndne_f32
            q = fminf(fmaxf(q, -QMAXF), QMAXF);
            y[k] = q * sv[k];
        }
        __builtin_nontemporal_store(y, &out[v]);           // streaming store (NT)
    }
}

// ---------------------------------------------------------------------------
// Launch
// inputs: d_in[0]=x (f32, 4*4096*4096), d_in[1]=group_scales (f32, 64),
//         d_in[2]=perm (i32, 4096), d_in[3]=inv_perm (i32, 4096)
// ---------------------------------------------------------------------------
extern "C" void kernel_launch(void* const* d_in, const int* in_sizes, int n_in,
                              void* d_out, int out_size, void* d_ws, size_t ws_size,
                              hipStream_t stream) {
    const float* x            = (const float*)d_in[0];
    const float* group_scales = (const float*)d_in[1];
    const int*   inv_perm     = (const int*)d_in[3];
    float*       out          = (float*)d_out;

    // workspace layout: [0..4095] scale LUT, [4096..8191] inverse-scale LUT
    float* scale_lut = (float*)d_ws;
    float* inv_lut   = scale_lut + CHN;

    build_scale_lut_kernel<<<CHN / 256, 256, 0, stream>>>(
        group_scales, inv_perm, scale_lut, inv_lut);

    long long nvec = (long long)out_size / 4;   // 16,777,216 float4s
    fakequant_kernel<<<4096, 256, 0, stream>>>(
        (const f4*)x, scale_lut, inv_lut, (f4*)out, nvec);
}